// GCNss_48593259987023
// MI455X (gfx1250) — compile-verified
//
#include <hip/hip_runtime.h>

typedef __attribute__((ext_vector_type(2))) float v2f;
typedef __attribute__((ext_vector_type(8))) float v8f;

#define N_NODES   100000
#define N_EDGES   3200000
#define N_FEAT    128
#define HIDDEN    128
#define N_CLASSES 10
#define N_GRAPHS  512
#define M_TILES   (N_NODES / 16)   // 6250, exact

// ---------------------------------------------------------------------------
// Kernel 1: dual GEMM via V_WMMA_F32_16X16X4_F32.
//   hrel = x @ W_rel
//   agg  = x @ W_root + b_rel     (initializes the scatter accumulation target)
// Block = 256 threads (8 waves). Both 128x128 f32 weight matrices are staged
// in LDS (128 KB; CDNA5 WGP has 320 KB) using the CDNA5 async memory->LDS
// path (GLOBAL_LOAD_ASYNC_TO_LDS_B128, tracked by ASYNCcnt) so the prologue
// never routes the weights through VGPRs. Each wave owns one 16-row M-tile,
// holds the full 16x128 A tile in registers, and sweeps the 8 N-tiles for
// both weight matrices, sharing the A fragments.
//
// WMMA f32 16x16x4 fragment layouts (ISA 7.12.2):
//   A (16x4):  lane l, vgpr v -> M = l%16, K = 2*(l/16) + v
//   B (4x16):  lane l, vgpr v -> N = l%16, K = 2*(l/16) + v
//   C/D(16x16):lane l, vgpr v -> N = l%16, M = v + 8*(l/16)
// ---------------------------------------------------------------------------
__global__ void gcn_dual_gemm_wmma(const float* __restrict__ x,
                                   const float* __restrict__ Wrel,
                                   const float* __restrict__ brel,
                                   const float* __restrict__ Wroot,
                                   float* __restrict__ hrel,
                                   float* __restrict__ agg) {
    extern __shared__ float smem[];            // 2 * 128*128 floats = 128 KB
    float* sWrel  = smem;
    float* sWroot = smem + N_FEAT * HIDDEN;

    // Async copy of both weight matrices into LDS (16 B per lane per issue).
    // Low 32 bits of a generic shared pointer are the wave-relative LDS byte
    // address (flat aperture rule).
    const int nvec = (N_FEAT * HIDDEN) / 4;    // 4096 float4 per matrix
    const float4* g0 = (const float4*)Wrel;
    const float4* g1 = (const float4*)Wroot;
    for (int i = threadIdx.x; i < nvec; i += blockDim.x) {
        unsigned lds0 = (unsigned)(size_t)(const void*)(((const float4*)sWrel)  + i);
        unsigned lds1 = (unsigned)(size_t)(const void*)(((const float4*)sWroot) + i);
        asm volatile("global_load_async_to_lds_b128 %0, %1, off"
                     :: "v"(lds0), "v"(g0 + i) : "memory");
        asm volatile("global_load_async_to_lds_b128 %0, %1, off"
                     :: "v"(lds1), "v"(g1 + i) : "memory");
    }
    asm volatile("s_wait_asynccnt 0x0" ::: "memory");
    __syncthreads();

    const int wave  = threadIdx.x >> 5;
    const int lane  = threadIdx.x & 31;
    const int mtile = blockIdx.x * 8 + wave;
    if (mtile >= M_TILES) return;              // uniform per wave: EXEC all-1s inside

    const int mrow  = (mtile << 4) + (lane & 15);
    const int khalf = (lane >> 4) << 1;        // 0 or 2
    const int ncol  = lane & 15;

    // Preload the full 16x128 A tile: 32 K-steps, one float2 per lane each.
    v2f afrag[32];
    const float* xrow = x + (size_t)mrow * N_FEAT;
#pragma unroll
    for (int ks = 0; ks < 32; ++ks) {
        afrag[ks] = *(const v2f*)(xrow + (ks << 2) + khalf);
    }

    for (int nt = 0; nt < 8; ++nt) {
        const int n0 = nt << 4;
        v8f crel  = {};
        v8f croot = {};
#pragma unroll
        for (int ks = 0; ks < 32; ++ks) {
            const int k0 = (ks << 2) + khalf;
            v2f bfr, bfo;
            bfr.x = sWrel [(k0    ) * HIDDEN + n0 + ncol];
            bfr.y = sWrel [(k0 + 1) * HIDDEN + n0 + ncol];
            bfo.x = sWroot[(k0    ) * HIDDEN + n0 + ncol];
            bfo.y = sWroot[(k0 + 1) * HIDDEN + n0 + ncol];
            crel  = __builtin_amdgcn_wmma_f32_16x16x4_f32(
                        false, afrag[ks], false, bfr, (short)0, crel,  false, false);
            croot = __builtin_amdgcn_wmma_f32_16x16x4_f32(
                        false, afrag[ks], false, bfo, (short)0, croot, false, false);
        }

        const float bias  = brel[n0 + ncol];
        const int   rbase = (mtile << 4) + ((lane >> 4) << 3);
#pragma unroll
        for (int v = 0; v < 8; ++v) {
            const size_t off = (size_t)(rbase + v) * HIDDEN + n0 + ncol;
            hrel[off] = crel[v];
            agg[off]  = croot[v] + bias;
        }
    }
}

// ---------------------------------------------------------------------------
// Kernel 2: edge scatter-add  agg[dst] += hrel[src].
// 32 lanes per edge; each lane moves a float4 (4 x global_atomic_add_f32).
// Both hrel and agg (102 MB total) fit in the 192 MB L2, so this stage is
// L2-atomic-throughput bound; f32 dword atomics are the minimal-op encoding.
// ---------------------------------------------------------------------------
__global__ void gcn_edge_scatter(const long long* __restrict__ src,
                                 const long long* __restrict__ dst,
                                 const float* __restrict__ hrel,
                                 float* __restrict__ agg) {
    const long long gid = (long long)blockIdx.x * blockDim.x + threadIdx.x;
    const long long e   = gid >> 5;
    if (e >= N_EDGES) return;
    const int f = ((int)(gid & 31)) << 2;

    const long long s = src[e];
    const long long d = dst[e];
    const float4 val = *(const float4*)(hrel + s * N_FEAT + f);
    float* p = agg + d * N_FEAT + f;
    atomicAdd(p + 0, val.x);
    atomicAdd(p + 1, val.y);
    atomicAdd(p + 2, val.z);
    atomicAdd(p + 3, val.w);
}

// ---------------------------------------------------------------------------
// Kernel 3: fused global_mean_pool + classifier.
// batch is sorted: each block (one graph) binary-searches its node range,
// does a coalesced per-feature mean into LDS, then 10 threads compute the
// 128-deep dot products against W_lin and write the output.
// ---------------------------------------------------------------------------
__global__ void gcn_pool_classify(const float* __restrict__ h,
                                  const long long* __restrict__ batch,
                                  const float* __restrict__ Wlin,
                                  const float* __restrict__ blin,
                                  float* __restrict__ out) {
    __shared__ float pooled[HIDDEN];
    const int g = blockIdx.x;
    const int f = threadIdx.x;                 // 128 threads

    // lower_bound(batch, g) and lower_bound(batch, g+1)
    int lo = 0, hi = N_NODES;
    while (lo < hi) { int mid = (lo + hi) >> 1; if (batch[mid] < (long long)g) lo = mid + 1; else hi = mid; }
    const int start = lo;
    hi = N_NODES;
    while (lo < hi) { int mid = (lo + hi) >> 1; if (batch[mid] < (long long)g + 1) lo = mid + 1; else hi = mid; }
    const int end = lo;

    float s = 0.0f;
    for (int i = start; i < end; ++i) s += h[(size_t)i * HIDDEN + f];
    const float cnt = (float)(end - start);
    pooled[f] = s / fmaxf(cnt, 1.0f);
    __syncthreads();

    if (f < N_CLASSES) {
        float acc = blin[f];
#pragma unroll 16
        for (int k = 0; k < HIDDEN; ++k) acc += pooled[k] * Wlin[k * N_CLASSES + f];
        out[g * N_CLASSES + f] = acc;
    }
}

// ---------------------------------------------------------------------------
extern "C" void kernel_launch(void* const* d_in, const int* in_sizes, int n_in,
                              void* d_out, int out_size, void* d_ws, size_t ws_size,
                              hipStream_t stream) {
    const float*     x     = (const float*)d_in[0];
    const long long* edge  = (const long long*)d_in[1];   // [2, N_EDGES]
    const long long* batch = (const long long*)d_in[2];
    const float*     Wrel  = (const float*)d_in[3];
    const float*     brel  = (const float*)d_in[4];
    const float*     Wroot = (const float*)d_in[5];
    const float*     Wlin  = (const float*)d_in[6];
    const float*     blin  = (const float*)d_in[7];
    float*           out   = (float*)d_out;

    float* hrel = (float*)d_ws;                              // 100000*128 f32 = 51.2 MB
    float* agg  = hrel + (size_t)N_NODES * HIDDEN;           // 51.2 MB more

    const long long* src = edge;
    const long long* dst = edge + N_EDGES;

    // Stage 1: dual GEMM (hrel, agg init). 128 KB dynamic LDS for both weights.
    const size_t shmem = 2ull * N_FEAT * HIDDEN * sizeof(float);
    const int gemm_blocks = (M_TILES + 7) / 8;               // 8 waves (M-tiles) / block
    hipLaunchKernelGGL(gcn_dual_gemm_wmma, dim3(gemm_blocks), dim3(256), shmem, stream,
                       x, Wrel, brel, Wroot, hrel, agg);

    // Stage 2: edge scatter-add.
    const long long threads = (long long)N_EDGES * 32;
    const int scatter_blocks = (int)((threads + 255) / 256);
    hipLaunchKernelGGL(gcn_edge_scatter, dim3(scatter_blocks), dim3(256), 0, stream,
                       src, dst, hrel, agg);

    // Stage 3: pooled mean + classifier.
    hipLaunchKernelGGL(gcn_pool_classify, dim3(N_GRAPHS), dim3(HIDDEN), 0, stream,
                       agg, batch, Wlin, blin, out);
}